// LearnedEvictionModel_77068893160340
// MI455X (gfx1250) — compile-verified
//
#include <hip/hip_runtime.h>
#include <hip/hip_bf16.h>

#define HIDDEN 64
#define VOCAB 64
#define SLOTS 8
#define SEQ 32
#define NSTEP 31  // SEQ - 1

typedef __attribute__((ext_vector_type(2))) float v2f;
typedef __attribute__((ext_vector_type(8))) float v8f;

// ---------------------------------------------------------------------------
// Kernel 1: per-token eviction score table.
// score[t] = ReLU(e_t @ sc_w1 + sc_b1) @ sc_w2 + sc_b2   (t = 0..63)
// Memory slots only ever hold exact embedding rows, so the scorer is a LUT.
// ---------------------------------------------------------------------------
__global__ void score_table_kernel(const float* __restrict__ embed,  // (66,64)
                                   const float* __restrict__ w1,     // (64,32)
                                   const float* __restrict__ b1,     // (32)
                                   const float* __restrict__ w2,     // (32)
                                   const float* __restrict__ b2,     // (1)
                                   float* __restrict__ scores)       // (64)
{
    const int t = threadIdx.x;
    if (t >= VOCAB) return;
    const float* e = embed + t * HIDDEN;
    float acc = b2[0];
    for (int j = 0; j < 32; ++j) {
        float h = b1[j];
        for (int i = 0; i < HIDDEN; ++i) h = fmaf(e[i], w1[i * 32 + j], h);
        acc = fmaf(fmaxf(h, 0.f), w2[j], acc);
    }
    scores[t] = acc;
}

// ---------------------------------------------------------------------------
// Kernel 2: sequential eviction scan as an integer token queue.
// One thread per batch; 8-slot queue fully unrolled in registers.
// Steps 0..7: fill. Steps 8..30: argmin(score[slot]) -> delete/shift/append.
// Emits the 8 final slot tokens packed into 8 bytes (uint2) per batch.
// ---------------------------------------------------------------------------
__global__ __launch_bounds__(256)
void scan_kernel(const int* __restrict__ seqs,       // (B,32)
                 const float* __restrict__ scores_g, // (64)
                 unsigned* __restrict__ slots_out,   // (B,2) u32
                 int B)
{
    __shared__ float sc[VOCAB];
    if (threadIdx.x < VOCAB) sc[threadIdx.x] = scores_g[threadIdx.x];
    __syncthreads();

    const int b = blockIdx.x * blockDim.x + threadIdx.x;
    if (b >= B) return;

    const int* s = seqs + b * SEQ;
    int slot[SLOTS];
#pragma unroll
    for (int i = 0; i < SLOTS; ++i) slot[i] = s[i] & 63;

    for (int t = SLOTS; t < NSTEP; ++t) {
        float best = sc[slot[0]];
        int bi = 0;
#pragma unroll
        for (int i = 1; i < SLOTS; ++i) {
            float v = sc[slot[i]];
            if (v < best) { best = v; bi = i; }  // strict '<' == first-min tie rule
        }
#pragma unroll
        for (int i = 0; i < SLOTS - 1; ++i)
            slot[i] = (i >= bi) ? slot[i + 1] : slot[i];
        slot[SLOTS - 1] = s[t] & 63;
    }

    unsigned px = (unsigned)slot[0] | ((unsigned)slot[1] << 8) |
                  ((unsigned)slot[2] << 16) | ((unsigned)slot[3] << 24);
    unsigned py = (unsigned)slot[4] | ((unsigned)slot[5] << 8) |
                  ((unsigned)slot[6] << 16) | ((unsigned)slot[7] << 24);
    slots_out[2 * b]     = px;
    slots_out[2 * b + 1] = py;
}

// ---------------------------------------------------------------------------
// Kernel 3: readout MLP with V_WMMA_F32_16X16X4_F32.
// h = [embed[q], mean(embed[slot_0..7])] (B x 128)
// out = ReLU(h @ rh_w1 + rh_b1) @ rh_w2 + rh_b2          (B x 64)
// 128 threads = 4 wave32; each wave owns one 16-row tile of h.
// ---------------------------------------------------------------------------
__global__ __launch_bounds__(128)
void readout_kernel(const int* __restrict__ qtok,       // (B)
                    const unsigned* __restrict__ slots, // (B,2)
                    const float* __restrict__ embed,    // (66,64) rows 0..63 used
                    const float* __restrict__ w1g,      // (128,64)
                    const float* __restrict__ b1g,      // (64)
                    const float* __restrict__ w2g,      // (64,64)
                    const float* __restrict__ b2g,      // (64)
                    float* __restrict__ out,            // (B,64)
                    int B)
{
    __shared__ float embLDS[64 * 64];   // 16 KB
    __shared__ float w1LDS[128 * 64];   // 32 KB
    __shared__ float w2LDS[64 * 64];    // 16 KB
    __shared__ float b1LDS[64];
    __shared__ float b2LDS[64];
    __shared__ float hbuf[4][16 * 128]; // 32 KB: per-wave h tile, reused as act
    __shared__ int      qts[4][16];
    __shared__ unsigned slts[4][16][2];

    const int tid  = threadIdx.x;
    const int wave = tid >> 5;
    const int lane = tid & 31;
    const int half = lane >> 4;  // 0: lanes 0-15, 1: lanes 16-31
    const int mrow = lane & 15;

    // Stage constants into LDS (whole block cooperates).
    for (int i = tid; i < 64 * 64; i += 128)  embLDS[i] = embed[i];
    for (int i = tid; i < 128 * 64; i += 128) w1LDS[i]  = w1g[i];
    for (int i = tid; i < 64 * 64; i += 128)  w2LDS[i]  = w2g[i];
    if (tid < 64) { b1LDS[tid] = b1g[tid]; b2LDS[tid] = b2g[tid]; }

    const int tile = blockIdx.x * 4 + wave;
    const int base = tile * 16;
    if (lane < 16) {
        qts[wave][lane]     = qtok[base + lane];
        slts[wave][lane][0] = slots[2 * (base + lane)];
        slts[wave][lane][1] = slots[2 * (base + lane) + 1];
    }
    __syncthreads();

    // Build this wave's 16x128 h tile in LDS.
    float* hw = hbuf[wave];
#pragma unroll 4
    for (int r = 0; r < 16; ++r) {
        const int qt = qts[wave][r];
        const unsigned sx = slts[wave][r][0];
        const unsigned sy = slts[wave][r][1];
        // query-embedding half (cols 0..63)
        hw[r * 128 + lane]      = embLDS[qt * 64 + lane];
        hw[r * 128 + 32 + lane] = embLDS[qt * 64 + 32 + lane];
        // memory-summary half (cols 64..127): mean of 8 slot embeddings
#pragma unroll
        for (int p = 0; p < 2; ++p) {
            const int cc = p * 32 + lane;
            float ssum = 0.f;
#pragma unroll
            for (int k = 0; k < 4; ++k) ssum += embLDS[((sx >> (8 * k)) & 63) * 64 + cc];
#pragma unroll
            for (int k = 0; k < 4; ++k) ssum += embLDS[((sy >> (8 * k)) & 63) * 64 + cc];
            hw[r * 128 + 64 + cc] = ssum * 0.125f;
        }
    }
    __syncthreads();

    // ---- Layer 1: (16x128) @ (128x64), K in steps of 4 via WMMA f32 ----
    // A frag (16x4): lane holds row mrow, K columns {4k+2*half, 4k+2*half+1}.
    // B frag (4x16): lane holds col mrow of the matching K rows.
    v8f c1[4];
#pragma unroll
    for (int n = 0; n < 4; ++n) {
        v8f c = {};
#pragma unroll
        for (int k = 0; k < 32; ++k) {
            const int k0 = 4 * k + 2 * half;
            v2f a, b;
            a.x = hw[mrow * 128 + k0];
            a.y = hw[mrow * 128 + k0 + 1];
            b.x = w1LDS[k0 * 64 + n * 16 + mrow];
            b.y = w1LDS[(k0 + 1) * 64 + n * 16 + mrow];
            c = __builtin_amdgcn_wmma_f32_16x16x4_f32(
                    false, a, false, b, (short)0, c, false, false);
        }
        c1[n] = c;
    }
    __syncthreads();  // all reads of hw done -> safe to overwrite with act

    // Bias + ReLU, write 16x64 activation tile (reuse hw).
    // D layout: element v -> (M = v + 8*half, N = n*16 + mrow).
#pragma unroll
    for (int n = 0; n < 4; ++n) {
        const float bn = b1LDS[n * 16 + mrow];
#pragma unroll
        for (int v = 0; v < 8; ++v)
            hw[(v + 8 * half) * 64 + n * 16 + mrow] = fmaxf(c1[n][v] + bn, 0.f);
    }
    __syncthreads();

    // ---- Layer 2: (16x64) @ (64x64) ----
#pragma unroll
    for (int n = 0; n < 4; ++n) {
        v8f c = {};
#pragma unroll
        for (int k = 0; k < 16; ++k) {
            const int k0 = 4 * k + 2 * half;
            v2f a, b;
            a.x = hw[mrow * 64 + k0];
            a.y = hw[mrow * 64 + k0 + 1];
            b.x = w2LDS[k0 * 64 + n * 16 + mrow];
            b.y = w2LDS[(k0 + 1) * 64 + n * 16 + mrow];
            c = __builtin_amdgcn_wmma_f32_16x16x4_f32(
                    false, a, false, b, (short)0, c, false, false);
        }
        const float bn = b2LDS[n * 16 + mrow];
#pragma unroll
        for (int v = 0; v < 8; ++v)
            out[(size_t)(base + v + 8 * half) * 64 + n * 16 + mrow] = c[v] + bn;
    }
}

// ---------------------------------------------------------------------------
extern "C" void kernel_launch(void* const* d_in, const int* in_sizes, int n_in,
                              void* d_out, int out_size, void* d_ws, size_t ws_size,
                              hipStream_t stream) {
    const int*   seqs  = (const int*)d_in[0];
    const int*   qtok  = (const int*)d_in[1];
    const float* embed = (const float*)d_in[2];
    const float* sc_w1 = (const float*)d_in[3];
    const float* sc_b1 = (const float*)d_in[4];
    const float* sc_w2 = (const float*)d_in[5];
    const float* sc_b2 = (const float*)d_in[6];
    const float* rh_w1 = (const float*)d_in[7];
    const float* rh_b1 = (const float*)d_in[8];
    const float* rh_w2 = (const float*)d_in[9];
    const float* rh_b2 = (const float*)d_in[10];
    float* out = (float*)d_out;

    const int B = in_sizes[1];  // query_tok has B elements

    // Workspace layout: [0,256): score LUT (64 f32). [256, 256+8*B): slot tokens.
    float*    scores = (float*)d_ws;
    unsigned* slots  = (unsigned*)((char*)d_ws + 256);

    score_table_kernel<<<1, 64, 0, stream>>>(embed, sc_w1, sc_b1, sc_w2, sc_b2, scores);
    scan_kernel<<<(B + 255) / 256, 256, 0, stream>>>(seqs, scores, slots, B);
    readout_kernel<<<B / 64, 128, 0, stream>>>(qtok, slots, embed,
                                               rh_w1, rh_b1, rh_w2, rh_b2, out, B);
}